// Conv2D_6124623364160
// MI455X (gfx1250) — compile-verified
//
#include <hip/hip_runtime.h>

typedef __attribute__((ext_vector_type(2))) float v2f;
typedef __attribute__((ext_vector_type(8))) float v8f;

#define IH 8192
#define IW 8192
#define OW 8191            // IW - KW + 1
#define TILES_X 511        // full 16-col output tiles: 511*16 = 8176
#define RUNS_PER_BAND 32
#define TILES_PER_RUN 16   // 32 runs * 16 tiles = 512 >= 511
#define TAIL_COL0 8176     // columns 8176..8190 handled by tail kernel
#define TAIL_N 15

// ---------------------------------------------------------------------------
// Bulk kernel: one wave computes a run of 16 consecutive 16x16 output tiles
// in a 16-row band via V_WMMA_F32_16X16X4_F32.
//   D(16x16) = sum_{kc=0..4} A_kc(16x4) x B_kc(4x16) + bias
//   A_kc[m,k] = x[row0+m, j0+4kc+k]   (data, ISA-documented A layout)
//   B_kc[k,n] = w0*[4kc+k==n] + w1*[4kc+k==n+1]  (weights, built in regs)
// ---------------------------------------------------------------------------
__global__ __launch_bounds__(256)
void conv1x2_wmma(const float* __restrict__ x, const float* __restrict__ wgt,
                  const float* __restrict__ bias, float* __restrict__ out) {
  const int lane = threadIdx.x & 31;
  const int waveId = blockIdx.x * 8 + (threadIdx.x >> 5);
  const int band = waveId >> 5;          // / RUNS_PER_BAND
  const int run  = waveId & 31;
  const int row0 = band << 4;
  const int m = lane & 15;               // A: matrix row ; D: column n
  const int h = lane >> 4;               // half-wave select

  const float w0 = wgt[0];
  const float w1 = wgt[1];
  const float bv = bias[0];

  // Build the 5 constant B fragments once per wave.
  // Slot (elem v, half h) of B holds B[k = v + 2h][n = m].
  v2f bfrag[5];
#pragma unroll
  for (int kc = 0; kc < 5; ++kc) {
#pragma unroll
    for (int v = 0; v < 2; ++v) {
      const int k = v + 2 * h;
      const int c = 4 * kc + k;          // x column relative to j0
      float val = 0.0f;
      if (c == m)     val += w0;
      if (c == m + 1) val += w1;
      bfrag[kc][v] = val;
    }
  }

  // A layout (32-bit 16x4): lane m = row; VGPR v, half h -> K = v + 2h.
  // So each lane loads a contiguous float2 at column j0 + 4kc + 2h.
  const float* rowPtr = x + (size_t)(row0 + m) * IW;

#pragma unroll 1
  for (int t = 0; t < TILES_PER_RUN; ++t) {
    const int tileCol = run * TILES_PER_RUN + t;
    if (tileCol >= TILES_X) break;       // wave-uniform: EXEC stays all-ones
    const int j0 = tileCol << 4;

    // Prefetch next tile's row segment into L0/L2 (global_prefetch_b8).
    __builtin_prefetch(rowPtr + j0 + 20 + 2 * h, 0, 1);

    v8f acc;
#pragma unroll
    for (int i = 0; i < 8; ++i) acc[i] = bv;   // C seeded with bias

#pragma unroll
    for (int kc = 0; kc < 5; ++kc) {
      const v2f a = *(const v2f*)(rowPtr + j0 + 4 * kc + 2 * h);
      // 8 args: (neg_a, A, neg_b, B, c_mod, C, reuse_a, reuse_b)
      acc = __builtin_amdgcn_wmma_f32_16x16x4_f32(
          false, a, false, bfrag[kc], (short)0, acc, false, false);
    }

    // D layout: VGPR r, half h -> M = r + 8h, N = m. Non-temporal stores:
    // the 256MB output stream must not evict the input stream from L2.
#pragma unroll
    for (int r = 0; r < 8; ++r) {
      const size_t o = (size_t)(row0 + r + 8 * h) * OW + (size_t)(j0 + m);
      __builtin_nontemporal_store(acc[r], out + o);
    }
  }
}

// ---------------------------------------------------------------------------
// Tail kernel: last 15 output columns (8176..8190), 8192 rows. Pure VALU.
// ---------------------------------------------------------------------------
__global__ __launch_bounds__(256)
void conv1x2_tail(const float* __restrict__ x, const float* __restrict__ wgt,
                  const float* __restrict__ bias, float* __restrict__ out) {
  const int idx = blockIdx.x * blockDim.x + threadIdx.x;
  const int total = IH * TAIL_N;
  if (idx >= total) return;
  const int i = idx / TAIL_N;
  const int j = TAIL_COL0 + (idx % TAIL_N);
  const float v = wgt[0] * x[(size_t)i * IW + j] +
                  wgt[1] * x[(size_t)i * IW + j + 1] + bias[0];
  __builtin_nontemporal_store(v, out + (size_t)i * OW + j);
}

extern "C" void kernel_launch(void* const* d_in, const int* in_sizes, int n_in,
                              void* d_out, int out_size, void* d_ws, size_t ws_size,
                              hipStream_t stream) {
  const float* x    = (const float*)d_in[0];   // 8192*8192 fp32
  const float* wgt  = (const float*)d_in[1];   // 1x2 fp32
  const float* bias = (const float*)d_in[2];   // 1 fp32
  float* out = (float*)d_out;                  // 8192*8191 fp32

  // 512 bands * 32 runs = 16384 waves; 8 waves (256 threads) per block.
  const int bulkBlocks = (512 * RUNS_PER_BAND) / 8;            // 2048
  conv1x2_wmma<<<bulkBlocks, 256, 0, stream>>>(x, wgt, bias, out);

  const int tailThreads = IH * TAIL_N;                          // 122880
  const int tailBlocks = (tailThreads + 255) / 256;             // 480
  conv1x2_tail<<<tailBlocks, 256, 0, stream>>>(x, wgt, bias, out);
}